// BatchedModelManifoldGeodesicFlow_37632503447572
// MI455X (gfx1250) — compile-verified
//
#include <hip/hip_runtime.h>
#include <math.h>

// ---------------------------------------------------------------------------
// Geometry: B=32, D=64, H=2048, O=64, n_steps=10 -> 256 interior points.
// corr_m = -0.5 * d/dz_m ||J(z)^T v||^2  (exact simplification of the
// Christoffel contraction; symmetric terms cancel under v (x) v).
// ---------------------------------------------------------------------------
#define DD    64
#define HH    2048
#define BB    32
#define NSTEP 10
#define NPTS  256      // (NSTEP-2)*BB
#define NSPLIT 8       // split-K factor for the K=2048 reduction GEMMs

typedef float v2f __attribute__((ext_vector_type(2)));
typedef float v8f __attribute__((ext_vector_type(8)));

// f32 WMMA 16x16x4. A fragment: lane m=L&15, hi=L>>4 holds A[m, kb+2*hi .. +1];
// B fragment mirrored with n=L&15; C/D: acc[r] is (M = r + 8*hi, N = L&15).
__device__ __forceinline__ v8f wmma_f32_k4(v2f a, v2f b, v8f c) {
    return __builtin_amdgcn_wmma_f32_16x16x4_f32(
        false, a, false, b, (short)0, c, false, false);
}

// ---------------------------------------------------------------------------
// K0a: v = (xT - x0) / ||xT - x0||  per batch row. 32 blocks x 32 threads.
// ---------------------------------------------------------------------------
__global__ void k_vnorm(const float* __restrict__ x0, const float* __restrict__ xT,
                        float* __restrict__ v) {
    int b = blockIdx.x, l = threadIdx.x;
    float d0 = xT[b * DD + l]      - x0[b * DD + l];
    float d1 = xT[b * DD + l + 32] - x0[b * DD + l + 32];
    float ss = d0 * d0 + d1 * d1;
    #pragma unroll
    for (int off = 16; off > 0; off >>= 1) ss += __shfl_xor(ss, off, 32);
    float inv = 1.0f / sqrtf(ss);
    v[b * DD + l]      = d0 * inv;
    v[b * DD + l + 32] = d1 * inv;
}

// ---------------------------------------------------------------------------
// K0b: straight paths -> out (all 10 steps), interior points -> Z [256 x 64].
// ---------------------------------------------------------------------------
__global__ void k_straight(const float* __restrict__ x0, const float* __restrict__ xT,
                           float* __restrict__ out, float* __restrict__ Z) {
    int idx = blockIdx.x * blockDim.x + threadIdx.x;
    if (idx >= NSTEP * BB * DD) return;
    int m   = idx & (DD - 1);
    int row = idx >> 6;            // s*32 + b
    int s   = row >> 5;
    int b   = row & 31;
    float a = x0[b * DD + m];
    float c = xT[b * DD + m];
    float t = (float)s * (1.0f / 9.0f);
    float val = a + t * (c - a);
    out[idx] = val;
    if (s >= 1 && s <= NSTEP - 2)
        Z[((s - 1) * BB + b) * DD + m] = val;
}

// ---------------------------------------------------------------------------
// Generic WMMA GEMM, C[M x N] = scale * A[M x K] * B[N x K]^T.
// One wave per 16x16 C tile. Launch exactly (M/16)*(N/16) waves.
// Used only for small-K (K=64) cases with plenty of tiles.
// ---------------------------------------------------------------------------
__global__ void k_gemm_abT(const float* __restrict__ A, int lda,
                           const float* __restrict__ B, int ldb,
                           float* __restrict__ C, int ldc,
                           int M, int N, int K, float scale) {
    int wave = (blockIdx.x * blockDim.x + threadIdx.x) >> 5;
    int lane = threadIdx.x & 31;
    int ntiles = N >> 4;
    int mt = wave / ntiles, nt = wave - mt * ntiles;
    if (mt >= (M >> 4)) return;                 // wave-uniform guard
    int m = lane & 15, hi = lane >> 4;
    const float* arow = A + (mt * 16 + m) * lda;
    const float* brow = B + (nt * 16 + m) * ldb;  // n = lane&15
    v8f acc = {0.f, 0.f, 0.f, 0.f, 0.f, 0.f, 0.f, 0.f};
    for (int kb = 0; kb < K; kb += 4) {
        int k0 = kb + 2 * hi;
        v2f a = {arow[k0], arow[k0 + 1]};
        v2f b = {brow[k0], brow[k0 + 1]};
        acc = wmma_f32_k4(a, b, acc);
    }
    int col = nt * 16 + m;
    #pragma unroll
    for (int r = 0; r < 8; ++r)
        C[(mt * 16 + r + 8 * hi) * ldc + col] = scale * acc[r];
}

// ---------------------------------------------------------------------------
// Split-K WMMA GEMM partials: PART[sp][M x N] = A[:, chunk_sp] * B[:, chunk_sp]^T.
// One wave per (split, 16x16 tile). Launch (M/16)*(N/16)*NSPLIT waves.
// Raises wave parallelism 8x for the K=2048, 256x64-output reductions.
// ---------------------------------------------------------------------------
__global__ void k_gemm_abT_sk(const float* __restrict__ A, int lda,
                              const float* __restrict__ B, int ldb,
                              float* __restrict__ PART,
                              int M, int N, int K) {
    int wave = (blockIdx.x * blockDim.x + threadIdx.x) >> 5;
    int lane = threadIdx.x & 31;
    int ntiles = N >> 4, mtiles = M >> 4;
    int tiles = ntiles * mtiles;
    int sp = wave / tiles;
    if (sp >= NSPLIT) return;                   // wave-uniform guard
    int t  = wave - sp * tiles;
    int mt = t / ntiles, nt = t - mt * ntiles;
    int kchunk = K / NSPLIT;
    int kbeg = sp * kchunk, kend = kbeg + kchunk;
    int m = lane & 15, hi = lane >> 4;
    const float* arow = A + (mt * 16 + m) * lda;
    const float* brow = B + (nt * 16 + m) * ldb;  // n = lane&15
    v8f acc = {0.f, 0.f, 0.f, 0.f, 0.f, 0.f, 0.f, 0.f};
    for (int kb = kbeg; kb < kend; kb += 4) {
        int k0 = kb + 2 * hi;
        v2f a = {arow[k0], arow[k0 + 1]};
        v2f b = {brow[k0], brow[k0 + 1]};
        acc = wmma_f32_k4(a, b, acc);
    }
    int col = nt * 16 + m;
    float* cpart = PART + sp * M * N;
    #pragma unroll
    for (int r = 0; r < 8; ++r)
        cpart[(mt * 16 + r + 8 * hi) * N + col] = acc[r];
}

// C[idx] = scale * sum_sp PART[sp][idx]  (deterministic split-K reduce)
__global__ void k_sk_reduce(const float* __restrict__ PART, float* __restrict__ C,
                            int MN, float scale) {
    int idx = blockIdx.x * blockDim.x + threadIdx.x;
    if (idx >= MN) return;
    float s = 0.f;
    #pragma unroll
    for (int sp = 0; sp < NSPLIT; ++sp) s += PART[sp * MN + idx];
    C[idx] = scale * s;
}

// ---------------------------------------------------------------------------
// K2: pre = Z*W1 + b1 ; t = tanh(pre); SW = (1-t^2)*w[b]; TSW = t*SW.
// M=256 (16 tiles) x N=2048 (128 tiles) -> 2048 waves.
// ---------------------------------------------------------------------------
__global__ void k_hidden(const float* __restrict__ Z,   // [256 x 64]
                         const float* __restrict__ W1,  // [64 x 2048]
                         const float* __restrict__ b1,  // [2048]
                         const float* __restrict__ w,   // [32 x 2048]
                         float* __restrict__ SW, float* __restrict__ TSW) {
    int wave = (blockIdx.x * blockDim.x + threadIdx.x) >> 5;
    int lane = threadIdx.x & 31;
    int mt = wave >> 7, nt = wave & 127;
    int m = lane & 15, hi = lane >> 4;
    const float* arow = Z + (mt * 16 + m) * DD;
    int n = nt * 16 + m;
    v8f acc = {0.f, 0.f, 0.f, 0.f, 0.f, 0.f, 0.f, 0.f};
    for (int kb = 0; kb < DD; kb += 4) {
        int k0 = kb + 2 * hi;
        v2f a = {arow[k0], arow[k0 + 1]};
        v2f b = {W1[k0 * HH + n], W1[(k0 + 1) * HH + n]};
        acc = wmma_f32_k4(a, b, acc);
    }
    float bias = b1[n];
    #pragma unroll
    for (int r = 0; r < 8; ++r) {
        int p = mt * 16 + r + 8 * hi;
        float t  = tanhf(acc[r] + bias);
        float s  = 1.0f - t * t;
        float sw = s * w[(p & 31) * HH + n];
        SW[p * HH + n]  = sw;
        TSW[p * HH + n] = t * sw;
    }
}

// ---------------------------------------------------------------------------
// K4: P = U*W1 ; Q = TSW .* P. Same tiling as k_hidden.
// ---------------------------------------------------------------------------
__global__ void k_pq(const float* __restrict__ U,    // [256 x 64]
                     const float* __restrict__ W1,   // [64 x 2048]
                     const float* __restrict__ TSW,  // [256 x 2048]
                     float* __restrict__ Q) {
    int wave = (blockIdx.x * blockDim.x + threadIdx.x) >> 5;
    int lane = threadIdx.x & 31;
    int mt = wave >> 7, nt = wave & 127;
    int m = lane & 15, hi = lane >> 4;
    const float* arow = U + (mt * 16 + m) * DD;
    int n = nt * 16 + m;
    v8f acc = {0.f, 0.f, 0.f, 0.f, 0.f, 0.f, 0.f, 0.f};
    for (int kb = 0; kb < DD; kb += 4) {
        int k0 = kb + 2 * hi;
        v2f a = {arow[k0], arow[k0 + 1]};
        v2f b = {W1[k0 * HH + n], W1[(k0 + 1) * HH + n]};
        acc = wmma_f32_k4(a, b, acc);
    }
    #pragma unroll
    for (int r = 0; r < 8; ++r) {
        int p = mt * 16 + r + 8 * hi;
        Q[p * HH + n] = TSW[p * HH + n] * acc[r];
    }
}

// ---------------------------------------------------------------------------
// K6: per interior point: scale = min(||corr||, 0.1); fac = t(1-t)*scale*0.1;
// out[interior row] += corr * fac.  One wave per point, 8 points/block.
// ---------------------------------------------------------------------------
__global__ void k_final(const float* __restrict__ CORR, float* __restrict__ out) {
    int p    = blockIdx.x * 8 + (threadIdx.x >> 5);
    int lane = threadIdx.x & 31;
    float c0 = CORR[p * DD + lane];
    float c1 = CORR[p * DD + lane + 32];
    float ss = c0 * c0 + c1 * c1;
    #pragma unroll
    for (int off = 16; off > 0; off >>= 1) ss += __shfl_xor(ss, off, 32);
    float nrm   = sqrtf(ss);
    float scale = fminf(nrm, 0.1f);
    int s_int = p >> 5, b = p & 31;
    float t   = (float)(s_int + 1) * (1.0f / 9.0f);
    float fac = t * (1.0f - t) * scale * 0.1f;
    int base = ((s_int + 1) * BB + b) * DD;
    out[base + lane]      += c0 * fac;
    out[base + lane + 32] += c1 * fac;
}

// ---------------------------------------------------------------------------
// inputs: x0, xT, W1, b1, W2, b2, n_steps  (b2 unused: J independent of b2)
// ---------------------------------------------------------------------------
extern "C" void kernel_launch(void* const* d_in, const int* in_sizes, int n_in,
                              void* d_out, int out_size, void* d_ws, size_t ws_size,
                              hipStream_t stream) {
    const float* x0 = (const float*)d_in[0];
    const float* xT = (const float*)d_in[1];
    const float* W1 = (const float*)d_in[2];
    const float* b1 = (const float*)d_in[3];
    const float* W2 = (const float*)d_in[4];
    float* out = (float*)d_out;

    float* ws   = (float*)d_ws;
    float* v    = ws;                // 32*64          = 2048
    float* wmat = v    + 2048;       // 32*2048        = 65536
    float* Z    = wmat + 65536;      // 256*64         = 16384
    float* SW   = Z    + 16384;      // 256*2048       = 524288
    float* TSW  = SW   + 524288;     // 256*2048       = 524288
    float* U    = TSW  + 524288;     // 256*64         = 16384
    float* CORR = U    + 16384;      // 256*64         = 16384
    float* PART = CORR + 16384;      // 8*256*64       = 131072
    float* Q    = SW;                // reuse: SW dead after U partials done

    // v and straight paths / interior Z
    k_vnorm   <<<32, 32, 0, stream>>>(x0, xT, v);
    k_straight<<<(NSTEP * BB * DD + 255) / 256, 256, 0, stream>>>(x0, xT, out, Z);

    // w = V[32x64] * W2[2048x64]^T  -> [32 x 2048]; tiles = 2*128 = 256 waves
    k_gemm_abT<<<32, 256, 0, stream>>>(v, DD, W2, DD, wmat, HH, BB, HH, DD, 1.0f);

    // hidden activations + SW/TSW (2048 waves)
    k_hidden<<<256, 256, 0, stream>>>(Z, W1, b1, wmat, SW, TSW);

    // U = SW[256x2048] * W1[64x2048]^T -> [256 x 64]; split-K: 64*8 = 512 waves
    k_gemm_abT_sk<<<64, 256, 0, stream>>>(SW, HH, W1, HH, PART, NPTS, DD, HH);
    k_sk_reduce  <<<64, 256, 0, stream>>>(PART, U, NPTS * DD, 1.0f);

    // Q = TSW .* (U * W1)   (2048 waves)
    k_pq<<<256, 256, 0, stream>>>(U, W1, TSW, Q);

    // corr = 2 * Q * W1^T -> [256 x 64]; split-K again
    k_gemm_abT_sk<<<64, 256, 0, stream>>>(Q, HH, W1, HH, PART, NPTS, DD, HH);
    k_sk_reduce  <<<64, 256, 0, stream>>>(PART, CORR, NPTS * DD, 2.0f);

    // clamp/scale and add into paths
    k_final<<<32, 256, 0, stream>>>(CORR, out);
}